// crime_prediction_weight_decay_80169859547453
// MI455X (gfx1250) — compile-verified
//
#include <hip/hip_runtime.h>

typedef float v2f __attribute__((ext_vector_type(2)));
typedef float v8f __attribute__((ext_vector_type(8)));

#define B_ 32
#define Q_ 32
#define S_ 1024
#define T_ 16
#define F_D 128
#define E_ 64
#define NH_ 16
#define INV_SQRT_DK 0.125f
#define RATIO_ 0.5f

// ---------------------------------------------------------------------------
// K1a: qt = query_t@tq_w + tq_b ; qt2[e'] = sum_e tk_w[e',e]*qt[e] ; cT = qt.tk_b
// ---------------------------------------------------------------------------
__global__ void proj_time(const float* __restrict__ query_t,
                          const float* __restrict__ tq_w, const float* __restrict__ tq_b,
                          const float* __restrict__ tk_w, const float* __restrict__ tk_b,
                          float* __restrict__ qt2, float* __restrict__ cT) {
  __shared__ float qtL[E_];
  __shared__ float red[E_];
  int bq = blockIdx.x;
  int e  = threadIdx.x;  // 0..63
  const float* xrow = query_t + (size_t)bq * E_;
  float acc = tq_b[e];
  for (int e0 = 0; e0 < E_; ++e0) acc += xrow[e0] * tq_w[e0 * E_ + e];
  qtL[e] = acc;
  __syncthreads();
  float acc2 = 0.f;
  const float* wrow = tk_w + e * E_;
  for (int k = 0; k < E_; ++k) acc2 += wrow[k] * qtL[k];
  qt2[(size_t)bq * E_ + e] = acc2;
  red[e] = qtL[e] * tk_b[e];
  __syncthreads();
  for (int off = 32; off > 0; off >>= 1) {
    if (e < off) red[e] += red[e + off];
    __syncthreads();
  }
  if (e == 0) cT[bq] = red[0];
}

// ---------------------------------------------------------------------------
// K1b: same for the position branch (per batch)
// ---------------------------------------------------------------------------
__global__ void proj_pos(const float* __restrict__ query_p,
                         const float* __restrict__ pq_w, const float* __restrict__ pq_b,
                         const float* __restrict__ pk_w, const float* __restrict__ pk_b,
                         float* __restrict__ qpw, float* __restrict__ cP) {
  __shared__ float qpL[E_];
  __shared__ float red[E_];
  int b = blockIdx.x;
  int e = threadIdx.x;
  const float* xrow = query_p + (size_t)b * E_;
  float acc = pq_b[e];
  for (int e0 = 0; e0 < E_; ++e0) acc += xrow[e0] * pq_w[e0 * E_ + e];
  qpL[e] = acc;
  __syncthreads();
  float acc2 = 0.f;
  const float* wrow = pk_w + e * E_;
  for (int k = 0; k < E_; ++k) acc2 += wrow[k] * qpL[k];
  qpw[(size_t)b * E_ + e] = acc2;
  red[e] = qpL[e] * pk_b[e];
  __syncthreads();
  for (int off = 32; off > 0; off >>= 1) {
    if (e < off) red[e] += red[e + off];
    __syncthreads();
  }
  if (e == 0) cP[b] = red[0];
}

// ---------------------------------------------------------------------------
// K2: st[b,q,s] = (qt2 . key_t[b,s] + cT)*inv_sqrt_dk ; softmax over s -> pt
// one block per (b,q), 256 threads, 4 s per thread
// ---------------------------------------------------------------------------
__global__ __launch_bounds__(256)
void time_softmax(const float* __restrict__ key_t, const float* __restrict__ qt2,
                  const float* __restrict__ cT, float* __restrict__ pt) {
  int b = blockIdx.x >> 5;
  int q = blockIdx.x & 31;
  __shared__ float qv[E_];
  __shared__ float red[256];
  int tid = threadIdx.x;
  if (tid < E_) qv[tid] = qt2[((size_t)(b * Q_ + q)) * E_ + tid];
  __syncthreads();
  float c = cT[b * Q_ + q];
  float st[4];
  float mx = -3.4e38f;
  for (int i = 0; i < 4; ++i) {
    int s = tid + i * 256;
    const float* krow = key_t + ((size_t)(b * S_ + s)) * E_;
    float acc = c;
    for (int e = 0; e < E_; ++e) acc += krow[e] * qv[e];
    st[i] = acc * INV_SQRT_DK;
    mx = fmaxf(mx, st[i]);
  }
  red[tid] = mx;
  __syncthreads();
  for (int off = 128; off > 0; off >>= 1) {
    if (tid < off) red[tid] = fmaxf(red[tid], red[tid + off]);
    __syncthreads();
  }
  mx = red[0];
  __syncthreads();
  float ex[4];
  float lsum = 0.f;
  for (int i = 0; i < 4; ++i) { ex[i] = __expf(st[i] - mx); lsum += ex[i]; }
  red[tid] = lsum;
  __syncthreads();
  for (int off = 128; off > 0; off >>= 1) {
    if (tid < off) red[tid] += red[tid + off];
    __syncthreads();
  }
  float inv = 1.0f / red[0];
  for (int i = 0; i < 4; ++i)
    pt[((size_t)(b * Q_ + q)) * S_ + tid + i * 256] = ex[i] * inv;
}

// ---------------------------------------------------------------------------
// K3: sp[b,s,t] = (key_p[b,s,t] . qpw + cP)*inv ; softmax over s -> pp[B,S,T]
// one block per (b,t)
// ---------------------------------------------------------------------------
__global__ __launch_bounds__(256)
void pos_softmax(const float* __restrict__ key_p, const float* __restrict__ qpw,
                 const float* __restrict__ cP, float* __restrict__ pp) {
  int b = blockIdx.x >> 4;
  int t = blockIdx.x & 15;
  __shared__ float qv[E_];
  __shared__ float red[256];
  int tid = threadIdx.x;
  if (tid < E_) qv[tid] = qpw[(size_t)b * E_ + tid];
  __syncthreads();
  float c = cP[b];
  float sp[4];
  float mx = -3.4e38f;
  for (int i = 0; i < 4; ++i) {
    int s = tid + i * 256;
    const float* krow = key_p + (((size_t)(b * S_ + s)) * T_ + t) * E_;
    float acc = c;
    for (int e = 0; e < E_; ++e) acc += krow[e] * qv[e];
    sp[i] = acc * INV_SQRT_DK;
    mx = fmaxf(mx, sp[i]);
  }
  red[tid] = mx;
  __syncthreads();
  for (int off = 128; off > 0; off >>= 1) {
    if (tid < off) red[tid] = fmaxf(red[tid], red[tid + off]);
    __syncthreads();
  }
  mx = red[0];
  __syncthreads();
  float ex[4];
  float lsum = 0.f;
  for (int i = 0; i < 4; ++i) { ex[i] = __expf(sp[i] - mx); lsum += ex[i]; }
  red[tid] = lsum;
  __syncthreads();
  for (int off = 128; off > 0; off >>= 1) {
    if (tid < off) red[tid] += red[tid + off];
    __syncthreads();
  }
  float inv = 1.0f / red[0];
  for (int i = 0; i < 4; ++i) {
    int s = tid + i * 256;
    pp[((size_t)(b * S_ + s)) * T_ + t] = ex[i] * inv;
  }
}

// ---------------------------------------------------------------------------
// K4 (hot): per (b,t): Y[q,f] = sum_s att[q,s]*value[b,s,t,f],  a[q] = sum_s att
// att[q,s] = 0.5*(pt[b,q,s]*twd[b,q,s,t] + pp[b,s,t]*lwd[b,q,s,t])
// 256 threads = 8 waves; wave (mt, nq): M-tile mt*16, N cols nq*32..+32
// S looped in chunks of 64; WMMA f32 16x16x4 over LDS-staged tiles.
// ---------------------------------------------------------------------------
#define VS_STRIDE 144  // 144 % 64 == 16 -> lane halves hit disjoint bank windows
#define AS_STRIDE 68   // 68 % 64 == 4  -> conflict-free A-frag rows, 8B aligned

__global__ __launch_bounds__(256)
void main_attn(const float* __restrict__ value,
               const float* __restrict__ twd, const float* __restrict__ lwd,
               const float* __restrict__ pt, const float* __restrict__ pp,
               float* __restrict__ Y, float* __restrict__ aOut) {
  int b = blockIdx.x >> 4;
  int t = blockIdx.x & 15;
  __shared__ float valS[64 * VS_STRIDE];
  __shared__ float attS[32 * AS_STRIDE];
  __shared__ float aPart[32][8];

  int tid  = threadIdx.x;
  int lane = tid & 31;
  int wave = tid >> 5;
  int m0 = (wave >> 2) * 16;
  int n0 = (wave & 3) * 32;

  v8f acc0 = {};
  v8f acc1 = {};
  float aAcc = 0.f;  // valid for tid<32 (q = tid)

  int aq  = tid >> 3;        // q row this thread fills in stage A
  int asb = (tid & 7) * 8;   // s sub-range base

  for (int c = 0; c < 16; ++c) {
    int s0 = c * 64;
    // ---- stage A: cooperative LDS fill ----
    for (int i = tid; i < 64 * 32; i += 256) {
      int row = i >> 5;
      int c4  = (i & 31) * 4;
      float4 v4 = *(const float4*)(value +
          (((size_t)(b * S_ + s0 + row)) * T_ + t) * F_D + c4);
      float* dst = &valS[row * VS_STRIDE + c4];
      dst[0] = v4.x; dst[1] = v4.y; dst[2] = v4.z; dst[3] = v4.w;
    }
    {
      const float* ptRow  = pt  + ((size_t)(b * Q_ + aq)) * S_ + s0 + asb;
      const float* twdRow = twd + (((size_t)(b * Q_ + aq)) * S_ + s0 + asb) * T_ + t;
      const float* lwdRow = lwd + (((size_t)(b * Q_ + aq)) * S_ + s0 + asb) * T_ + t;
      const float* ppCol  = pp  + ((size_t)(b * S_ + s0 + asb)) * T_ + t;
      float psum = 0.f;
      for (int j = 0; j < 8; ++j) {
        float a = RATIO_ * (ptRow[j] * twdRow[j * T_]) +
                  (1.0f - RATIO_) * (ppCol[j * T_] * lwdRow[j * T_]);
        attS[aq * AS_STRIDE + asb + j] = a;
        psum += a;
      }
      aPart[aq][tid & 7] = psum;
    }
    __syncthreads();
    if (tid < 32) {
      float s = 0.f;
      for (int j = 0; j < 8; ++j) s += aPart[tid][j];
      aAcc += s;
    }
    // ---- stage B: WMMA over the 64-wide S chunk ----
    int mrow = m0 + (lane & 15);
    int h2   = (lane >> 4) * 2;
    int ncol = lane & 15;
#pragma unroll
    for (int k = 0; k < 16; ++k) {
      int kk = k * 4 + h2;
      float2 av = *(const float2*)&attS[mrow * AS_STRIDE + kk];
      v2f a;  a.x = av.x;  a.y = av.y;
      v2f b0, b1;
      b0.x = valS[kk * VS_STRIDE + n0 + ncol];
      b0.y = valS[(kk + 1) * VS_STRIDE + n0 + ncol];
      b1.x = valS[kk * VS_STRIDE + n0 + 16 + ncol];
      b1.y = valS[(kk + 1) * VS_STRIDE + n0 + 16 + ncol];
      acc0 = __builtin_amdgcn_wmma_f32_16x16x4_f32(false, a, false, b0,
                                                   (short)0, acc0, false, false);
      acc1 = __builtin_amdgcn_wmma_f32_16x16x4_f32(false, a, false, b1,
                                                   (short)0, acc1, false, false);
    }
    __syncthreads();
  }
  // ---- write out D tiles: VGPR r -> M = r (+8 for upper lane half) ----
  int h  = lane >> 4;
  int nn = lane & 15;
#pragma unroll
  for (int r = 0; r < 8; ++r) {
    int m = m0 + r + 8 * h;
    size_t base = (((size_t)(b * Q_ + m)) * T_ + t) * F_D;
    Y[base + n0 + nn]      = acc0[r];
    Y[base + n0 + 16 + nn] = acc1[r];
  }
  if (tid < 32) aOut[((size_t)(b * Q_ + tid)) * T_ + t] = aAcc;
}

// ---------------------------------------------------------------------------
// K5: per (b,q): x[t,f] = Y[t,:]@val_w + a[t]*val_b ; flat view [T,F]->[F,T];
// merge over T with merge_w; then @nt_w + nt_b -> out[b,q,NH]
// ---------------------------------------------------------------------------
#define YS_STRIDE 132

__global__ __launch_bounds__(256)
void final_merge(const float* __restrict__ Y, const float* __restrict__ aBQT,
                 const float* __restrict__ val_w, const float* __restrict__ val_b,
                 const float* __restrict__ merge_w, const float* __restrict__ merge_b,
                 const float* __restrict__ nt_w, const float* __restrict__ nt_b,
                 float* __restrict__ out) {
  int bq = blockIdx.x;
  __shared__ float yS[16 * YS_STRIDE];
  __shared__ float aS[16];
  __shared__ float xS[16][129];
  __shared__ float x3S[128];
  int tid  = threadIdx.x;
  int lane = tid & 31;
  int wave = tid >> 5;
  int n0   = wave * 16;

  for (int i = tid; i < 16 * 32; i += 256) {
    int row = i >> 5;
    int c4  = (i & 31) * 4;
    float4 v4 = *(const float4*)(Y + ((size_t)bq * T_ + row) * F_D + c4);
    float* dst = &yS[row * YS_STRIDE + c4];
    dst[0] = v4.x; dst[1] = v4.y; dst[2] = v4.z; dst[3] = v4.w;
  }
  if (tid < 16) aS[tid] = aBQT[(size_t)bq * T_ + tid];
  __syncthreads();

  v8f acc = {};
  int m  = lane & 15;
  int h2 = (lane >> 4) * 2;
  int nn = lane & 15;
#pragma unroll 4
  for (int k = 0; k < 32; ++k) {
    int kk = k * 4 + h2;
    float2 av = *(const float2*)&yS[m * YS_STRIDE + kk];
    v2f a;  a.x = av.x;  a.y = av.y;
    v2f bw;
    bw.x = val_w[kk * F_D + n0 + nn];
    bw.y = val_w[(kk + 1) * F_D + n0 + nn];
    acc = __builtin_amdgcn_wmma_f32_16x16x4_f32(false, a, false, bw,
                                                (short)0, acc, false, false);
  }
  float vb = val_b[n0 + nn];
  int h = lane >> 4;
#pragma unroll
  for (int r = 0; r < 8; ++r) {
    int tt = r + 8 * h;
    xS[tt][n0 + nn] = acc[r] + aS[tt] * vb;
  }
  __syncthreads();
  // faithful reshape [T,F]->[F,T]: x3[i] = mb + sum_j flat[i*16+j]*mw[j]
  if (tid < 128) {
    int i = tid;
    int trow  = i >> 3;
    int fbase = (i * 16) & 127;
    float s = merge_b[0];
    for (int j = 0; j < 16; ++j) s += xS[trow][fbase + j] * merge_w[j];
    x3S[i] = s;
  }
  __syncthreads();
  if (tid < NH_) {
    float s = nt_b[tid];
    for (int i = 0; i < 128; ++i) s += x3S[i] * nt_w[i * NH_ + tid];
    out[(size_t)bq * NH_ + tid] = s;
  }
}

// ---------------------------------------------------------------------------
extern "C" void kernel_launch(void* const* d_in, const int* in_sizes, int n_in,
                              void* d_out, int out_size, void* d_ws, size_t ws_size,
                              hipStream_t stream) {
  (void)in_sizes; (void)n_in; (void)out_size; (void)ws_size;
  const float* query_t = (const float*)d_in[0];
  const float* key_t   = (const float*)d_in[1];
  const float* query_p = (const float*)d_in[2];
  const float* key_p   = (const float*)d_in[3];
  const float* value   = (const float*)d_in[4];
  const float* twd     = (const float*)d_in[5];
  const float* lwd     = (const float*)d_in[6];
  const float* tq_w    = (const float*)d_in[7];
  const float* tq_b    = (const float*)d_in[8];
  const float* tk_w    = (const float*)d_in[9];
  const float* tk_b    = (const float*)d_in[10];
  const float* pq_w    = (const float*)d_in[11];
  const float* pq_b    = (const float*)d_in[12];
  const float* pk_w    = (const float*)d_in[13];
  const float* pk_b    = (const float*)d_in[14];
  const float* val_w   = (const float*)d_in[15];
  const float* val_b   = (const float*)d_in[16];
  const float* merge_w = (const float*)d_in[17];
  const float* merge_b = (const float*)d_in[18];
  const float* nt_w    = (const float*)d_in[19];
  const float* nt_b    = (const float*)d_in[20];
  float* out = (float*)d_out;

  float* ws  = (float*)d_ws;
  float* pt  = ws;                                  // B*Q*S
  float* pp  = pt  + (size_t)B_ * Q_ * S_;          // B*S*T
  float* qt2 = pp  + (size_t)B_ * S_ * T_;          // B*Q*E
  float* cT  = qt2 + (size_t)B_ * Q_ * E_;          // B*Q
  float* qpw = cT  + (size_t)B_ * Q_;               // B*E
  float* cP  = qpw + (size_t)B_ * E_;               // B
  float* Yb  = cP  + (size_t)B_;                    // B*Q*T*F
  float* aB  = Yb  + (size_t)B_ * Q_ * T_ * F_D;    // B*Q*T

  proj_time  <<<B_ * Q_, 64,  0, stream>>>(query_t, tq_w, tq_b, tk_w, tk_b, qt2, cT);
  proj_pos   <<<B_,      64,  0, stream>>>(query_p, pq_w, pq_b, pk_w, pk_b, qpw, cP);
  time_softmax<<<B_ * Q_, 256, 0, stream>>>(key_t, qt2, cT, pt);
  pos_softmax<<<B_ * T_, 256, 0, stream>>>(key_p, qpw, cP, pp);
  main_attn  <<<B_ * T_, 256, 0, stream>>>(value, twd, lwd, pt, pp, Yb, aB);
  final_merge<<<B_ * Q_, 256, 0, stream>>>(Yb, aB, val_w, val_b, merge_w, merge_b,
                                           nt_w, nt_b, out);
}